// MammalContrastiveCriterion_88064009437767
// MI455X (gfx1250) — compile-verified
//
#include <hip/hip_runtime.h>
#include <math.h>

typedef __attribute__((ext_vector_type(2))) float v2f;
typedef __attribute__((ext_vector_type(8))) float v8f;

#define WAVES_PER_BLOCK 4
#define ROWS_PER_WAVE   16
#define ROWS_PER_BLOCK  (WAVES_PER_BLOCK * ROWS_PER_WAVE)   // 64
#define TK              64                                  // K-tile (floats)
#define ROW_STRIDE      (TK + 4)                            // 68: conflict-free banking
#define PAIR_BYTES      (2 * ROW_STRIDE * 4)                // 544: 2 rows per async instr
#define BUF_BYTES       (ROWS_PER_BLOCK * ROW_STRIDE * 4)   // 17408 per buffer

// ---------------------------------------------------------------------------
// Kernel 1: normalize text_concept rows (C=4) into a 16-row, zero-padded
// workspace so the WMMA B-operand columns 4..15 are exactly zero.
// ---------------------------------------------------------------------------
__global__ void prep_tn_kernel(const float* __restrict__ tc,
                               float* __restrict__ tnws,
                               int C, int Dd) {
    for (int i = threadIdx.x; i < (16 - C) * Dd; i += blockDim.x)
        tnws[C * Dd + i] = 0.0f;

    const int wave = threadIdx.x >> 5;
    const int lane = threadIdx.x & 31;
    if (wave < C) {
        const float* row = tc + (size_t)wave * Dd;
        float ss = 0.0f;
        for (int i = lane; i < Dd; i += 32) {
            float v = row[i];
            ss += v * v;
        }
        for (int off = 16; off > 0; off >>= 1)
            ss += __shfl_down(ss, off, 32);
        ss = __shfl(ss, 0, 32);
        const float inv = 1.0f / fmaxf(sqrtf(ss), 1e-12f);
        for (int i = lane; i < Dd; i += 32)
            tnws[(size_t)wave * Dd + i] = row[i] * inv;
    }
}

// ---------------------------------------------------------------------------
// Kernel 2: main fused pass.  Each wave owns 16 rows of x.
//   - x tiles staged to LDS with global_load_async_to_lds_b128 (coalesced,
//     ASYNCcnt-tracked, double-buffered per wave -> no block barriers)
//   - sim_raw accumulated with v_wmma_f32_16x16x4_f32 (A from LDS, B = padded
//     normalized-T from the L2-resident workspace)
//   - ||x_m||^2 via VALU FMAs co-executing with the WMMAs
//   - per-row 4-way softmax, block partial sum.
// ---------------------------------------------------------------------------
__global__ void sim_softmax_kernel(const float* __restrict__ x,
                                   const float* __restrict__ tnws,
                                   const int* __restrict__ flagp,
                                   float* __restrict__ partials,
                                   int Dd) {
    __shared__ __align__(16) float xbuf[2][ROWS_PER_BLOCK][ROW_STRIDE];

    const int lane    = threadIdx.x & 31;
    const int wave    = threadIdx.x >> 5;
    const int rowBase = (int)blockIdx.x * ROWS_PER_BLOCK + wave * ROWS_PER_WAVE;
    const int m       = lane & 15;          // WMMA A row / B col for this lane
    const int koff    = (lane >> 4) * 2;    // K sub-offset within each 4-chunk

    // ---- async staging addressing: one b128 instruction moves 2 rows ----
    const int sr = lane >> 4;               // which row of the pair
    const int sc = (lane & 15) * 4;         // column in floats (16B per lane)
    const float* gsrc = x + (size_t)(rowBase + sr) * Dd + sc;
    const unsigned lbase =
        (unsigned)(unsigned long long)&xbuf[0][wave * ROWS_PER_WAVE + sr][sc];

    const int NT = Dd / TK;                 // 16 tiles of K

    // stage tile t into buffer b (8 async b128 per wave; rows are wave-private)
    auto stage = [&](int t, int b) {
        const unsigned lb = lbase + (unsigned)b * (unsigned)BUF_BYTES;
        const float*   g  = gsrc + (size_t)t * TK;
        #pragma unroll
        for (int i = 0; i < 8; ++i) {
            asm volatile("global_load_async_to_lds_b128 %0, %1, off"
                         :: "v"(lb + (unsigned)(i * PAIR_BYTES)),
                            "v"((unsigned long long)(g + (size_t)(2 * i) * Dd))
                         : "memory");
        }
    };

    v8f   csim = {};
    float nacc = 0.0f;

    const float* trow = tnws + (size_t)m * Dd + koff;   // B operand source
    const float* lrow = &xbuf[0][wave * ROWS_PER_WAVE + m][koff];

    stage(0, 0);
    for (int t = 0; t < NT; ++t) {
        if (t + 1 < NT) {
            stage(t + 1, (t + 1) & 1);
            // current tile's 8 loads complete when <= 8 remain outstanding
            asm volatile("s_wait_asynccnt 0x8" ::: "memory");
        } else {
            asm volatile("s_wait_asynccnt 0x0" ::: "memory");
        }

        const float* la = lrow + (size_t)(t & 1) * (BUF_BYTES / 4);
        const float* tb = trow + (size_t)t * TK;
        #pragma unroll
        for (int kl = 0; kl < TK; kl += 4) {
            v2f a = *(const v2f*)(la + kl);     // ds_load_b64 (conflict-free)
            v2f b = *(const v2f*)(tb + kl);     // global_load_b64 (L2-resident)
            csim = __builtin_amdgcn_wmma_f32_16x16x4_f32(
                false, a, false, b, (short)0, csim, false, false);
            nacc = fmaf(a.x, a.x, nacc);
            nacc = fmaf(a.y, a.y, nacc);
        }
    }

    const int flag = *flagp;
    float local = 0.0f;

    // Accumulator layout: component r, lanes 0-15 -> row r (N = lane),
    // lanes 16-31 -> row r+8 (N = lane-16).
    // Row m's squared norm = nacc[lane m] + nacc[lane m+16].
    #pragma unroll
    for (int r = 0; r < 8; ++r) {
        float s  = csim[r];
        float s0 = __shfl(s, 0, 32), s1 = __shfl(s, 1, 32);
        float s2 = __shfl(s, 2, 32), s3 = __shfl(s, 3, 32);
        float t0 = __shfl(s, 16, 32), t1 = __shfl(s, 17, 32);
        float t2 = __shfl(s, 18, 32), t3 = __shfl(s, 19, 32);
        float n2a = __shfl(nacc, r, 32)     + __shfl(nacc, r + 16, 32);
        float n2b = __shfl(nacc, r + 8, 32) + __shfl(nacc, r + 24, 32);
        if (lane == 0) {
            {
                float inv = 1.0f / fmaxf(sqrtf(n2a), 1e-12f);
                float c0 = s0 * inv, c1 = s1 * inv, c2 = s2 * inv, c3 = s3 * inv;
                float mx = fmaxf(fmaxf(c0, c1), fmaxf(c2, c3));
                float e0 = expf(c0 - mx), e1 = expf(c1 - mx);
                float e2 = expf(c2 - mx), e3 = expf(c3 - mx);
                float sel = (flag == 0) ? e0 : (flag == 1) ? e1 : (flag == 2) ? e2 : e3;
                local += sel / (e0 + e1 + e2 + e3);
            }
            {
                float inv = 1.0f / fmaxf(sqrtf(n2b), 1e-12f);
                float c0 = t0 * inv, c1 = t1 * inv, c2 = t2 * inv, c3 = t3 * inv;
                float mx = fmaxf(fmaxf(c0, c1), fmaxf(c2, c3));
                float e0 = expf(c0 - mx), e1 = expf(c1 - mx);
                float e2 = expf(c2 - mx), e3 = expf(c3 - mx);
                float sel = (flag == 0) ? e0 : (flag == 1) ? e1 : (flag == 2) ? e2 : e3;
                local += sel / (e0 + e1 + e2 + e3);
            }
        }
    }

    __shared__ float wpart[WAVES_PER_BLOCK];
    if (lane == 0) wpart[wave] = local;
    __syncthreads();
    if (threadIdx.x == 0) {
        float s = 0.0f;
        #pragma unroll
        for (int i = 0; i < WAVES_PER_BLOCK; ++i) s += wpart[i];
        partials[blockIdx.x] = s;
    }
}

// ---------------------------------------------------------------------------
// Kernel 3: deterministic finalize: out = sum(partials) / B
// ---------------------------------------------------------------------------
__global__ void finalize_kernel(const float* __restrict__ partials,
                                int n, float invB,
                                float* __restrict__ out) {
    const int lane = threadIdx.x;   // launched with 32 threads
    float s = 0.0f;
    for (int i = lane; i < n; i += 32) s += partials[i];
    for (int off = 16; off > 0; off >>= 1) s += __shfl_down(s, off, 32);
    if (lane == 0) out[0] = s * invB;
}

extern "C" void kernel_launch(void* const* d_in, const int* in_sizes, int n_in,
                              void* d_out, int out_size, void* d_ws, size_t ws_size,
                              hipStream_t stream) {
    const float* x     = (const float*)d_in[0];
    // d_in[1] = label (unused by the reference computation)
    const float* tc    = (const float*)d_in[2];
    const int*   flagp = (const int*)d_in[3];

    const int C  = 4;
    const int Dd = in_sizes[2] / C;      // 1024
    const int B  = in_sizes[0] / Dd;     // 65536

    float* tnws     = (float*)d_ws;      // 16 * Dd floats (64 KB)
    float* partials = tnws + 16 * Dd;    // nBlocks floats

    const int nBlocks = B / ROWS_PER_BLOCK;   // 1024

    prep_tn_kernel<<<1, 256, 0, stream>>>(tc, tnws, C, Dd);
    sim_softmax_kernel<<<nBlocks, WAVES_PER_BLOCK * 32, 0, stream>>>(
        x, tnws, flagp, partials, Dd);
    finalize_kernel<<<1, 32, 0, stream>>>(partials, nBlocks, 1.0f / (float)B,
                                          (float*)d_out);
}